// Torsion_5119601016910
// MI455X (gfx1250) — compile-verified
//
#include <hip/hip_runtime.h>
#include <hip/hip_bf16.h>

#define NUM_SPH 7
#define NUM_RAD 6
#define NUM_EDGES 250000
#define NUM_TRIPLETS 500000
#define NCBF 49      // 7*7 spherical harmonics
#define KPAD 52      // 49 monomials padded to multiple of 4 (WMMA K)
#define NPAD 64      // 49 cbf columns padded to 4 x 16 (WMMA N tiles)
#define OUT_COLS 294 // 7*7*6

typedef __attribute__((ext_vector_type(2))) float v2f;
typedef __attribute__((ext_vector_type(8))) float v8f;

// Spherical Bessel zeros Z[l][i] (roots i=1..6 of j_l), float32 like the reference.
__device__ __constant__ float Zc[7][6] = {
  { 3.14159265358979f,  6.28318530717959f,  9.42477796076938f, 12.56637061435917f, 15.70796326794897f, 18.84955592153876f },
  { 4.49340945790906f,  7.72525183693771f, 10.90412165942890f, 14.06619391283147f, 17.22075527193077f, 20.37130295928756f },
  { 5.76345919689455f,  9.09501133047636f, 12.32294097056658f, 15.51460301088675f, 18.68903635536282f, 21.85387422270933f },
  { 6.98793200050052f, 10.41711854737937f, 13.69802315324925f, 16.92362128521432f, 20.12180617445461f, 23.30424698893965f },
  { 8.18256145257124f, 11.70490715457130f, 15.03966470761682f, 18.30125595954180f, 21.52541773313980f, 24.72756554783503f },
  { 9.35581211104275f, 12.96653017277532f, 16.35470963664259f, 19.65315210182132f, 22.90455064790271f, 26.12775013722569f },
  {10.51283540809400f, 14.20739245884246f, 17.64797483234651f, 20.98346306895834f, 24.26276800843056f, 27.50786815156354f }
};

// K[l][m] = sqrt((2l+1)/(4pi) * (l-m)!/(l+m)!)
__device__ __constant__ float Kc[7][7] = {
  { 0.2820947917738781f, 0, 0, 0, 0, 0, 0 },
  { 0.4886025119029199f, 0.3454941494713355f, 0, 0, 0, 0, 0 },
  { 0.6307831305050401f, 0.2575161346821265f, 0.1287580673410633f, 0, 0, 0, 0 },
  { 0.7463526651802308f, 0.2154534560761005f, 0.0681323650955524f, 0.0278149215755189f, 0, 0, 0 },
  { 0.8462843753216345f, 0.1892349391515121f, 0.0446031029038193f, 0.0119206806752225f, 0.0042145970709046f, 0, 0 },
  { 0.9356025796273889f, 0.1708168792406446f, 0.0322813618391345f, 0.0065894163133449f, 0.0015531731139135f, 0.0004911661854444f, 0 },
  { 1.0171072362820548f, 0.1569420096408554f, 0.0248150768326461f, 0.0041358196105536f, 0.0007550913435268f, 0.0001609860269925f, 0.0000464727381944f }
};

// spherical Bessel j_l(z) by upward recurrence (matches reference _sph_jn)
__device__ __forceinline__ float sph_jn(int l, float z) {
  float s = sinf(z), c = cosf(z);
  float inv = 1.0f / z;
  float j0 = s * inv;
  if (l == 0) return j0;
  float j1 = (s * inv - c) * inv;
  float jp = j0, jc = j1;
  #pragma unroll
  for (int i = 2; i <= l; ++i) {
    float jn = (float)(2 * i - 1) * inv * jc - jp;
    jp = jc; jc = jn;
  }
  return jc;
}

// ---------------------------------------------------------------------------
// Init kernel: build (a) Wt[NPAD][KPAD]  — cbf-column-major coefficient matrix
// mapping trig monomials -> real spherical harmonics, (b) norm[42] Bessel norms.
// Runs every launch (d_ws is poisoned by the harness). Single block.
// ---------------------------------------------------------------------------
__global__ void torsion_init(float* __restrict__ ws) {
  float* Wt   = ws;               // [NPAD][KPAD], Wt[col*KPAD + monomialRow]
  float* norm = ws + NPAD * KPAD; // [42]
  const int tid = threadIdx.x;

  for (int i = tid; i < NPAD * KPAD; i += blockDim.x) Wt[i] = 0.0f;
  __syncthreads();

  if (tid == 0) {
    // P_l^m(ct) = st^m * sum_d coef[l][m][d] * ct^d  (same recurrences as reference)
    double coef[7][7][7] = {};
    coef[0][0][0] = 1.0;
    for (int m = 1; m <= 6; ++m) coef[m][m][0] = -(2.0 * m - 1.0) * coef[m-1][m-1][0];
    for (int m = 0; m <= 5; ++m) coef[m+1][m][1] = (2.0 * m + 1.0) * coef[m][m][0];
    for (int m = 0; m <= 6; ++m)
      for (int l = m + 2; l <= 6; ++l)
        for (int d = 0; d <= 6; ++d) {
          double v = -(double)(l + m - 1) * coef[l-2][m][d];
          if (d >= 1) v += (double)(2 * l - 1) * coef[l-1][m][d-1];
          coef[l][m][d] = v / (double)(l - m);
        }
    const double SQRT2 = 1.4142135623730951;
    int row = 0;
    for (int m = 0; m <= 6; ++m) {
      for (int d = 0; d + m <= 6; ++d) {
        // cos-type monomial: cos(m*phi) * st^m * ct^d  (m==0 -> plain)
        for (int l = m; l <= 6; ++l) {
          double w = (m ? SQRT2 : 1.0) * (double)Kc[l][m] * coef[l][m][d];
          Wt[(l * l + l + m) * KPAD + row] = (float)w;
        }
        ++row;
        if (m > 0) {
          // sin-type monomial: sin(m*phi) * st^m * ct^d
          for (int l = m; l <= 6; ++l) {
            double w = SQRT2 * (double)Kc[l][m] * coef[l][m][d];
            Wt[(l * l + l - m) * KPAD + row] = (float)w;
          }
          ++row;
        }
      }
    }
  }

  if (tid < 42) {
    int l = tid / 6, i = tid % 6;
    float j = sph_jn(l + 1, Zc[l][i]);
    norm[tid] = rsqrtf(0.5f * j * j); // 1/sqrt(0.5*j_{l+1}(z)^2)
  }
}

// ---------------------------------------------------------------------------
// Main fused kernel: 1 triplet per lane, 32 triplets per wave, 8 waves/block.
// cbf = monomials x W via V_WMMA_F32_16X16X4_F32 (two 16-row tiles per wave),
// rbf recomputed in-register from dist[idx] (L2-resident), NT streaming stores.
// ---------------------------------------------------------------------------
__global__ void __launch_bounds__(256)
torsion_main(const float* __restrict__ dist,
             const float* __restrict__ angle,
             const float* __restrict__ phiArr,
             const int*   __restrict__ idx_kj,
             const float* __restrict__ ws,
             float* __restrict__ out)
{
  __shared__ float smem[8 * 32 * KPAD]; // 53 KB: per-wave mono -> cbf staging
  const int lane = threadIdx.x & 31;
  const int wid  = threadIdx.x >> 5;
  const int hi   = (lane >> 4) & 1;  // lane half (WMMA K split)
  const int l16  = lane & 15;
  float* buf = smem + wid * 32 * KPAD;

  const int t  = blockIdx.x * 256 + threadIdx.x;
  const int tc = (t < NUM_TRIPLETS) ? t : (NUM_TRIPLETS - 1); // clamp: EXEC stays all-1s

  // ---- per-lane trig monomials into LDS (A-matrix rows) ----
  const float th = angle[tc], ph = phiArr[tc];
  const float ct = cosf(th), st = sinf(th);
  const float cp = cosf(ph), sp = sinf(ph);
  {
    float* mrow = buf + lane * KPAD;
    float cm = 1.0f, sm = 0.0f, stm = 1.0f;
    int row = 0;
    #pragma unroll
    for (int m = 0; m <= 6; ++m) {
      if (m > 0) {
        float cn = cm * cp - sm * sp;
        float sn = sm * cp + cm * sp;
        cm = cn; sm = sn;
        stm *= st;
      }
      float ctd = 1.0f;
      #pragma unroll
      for (int d = 0; d + m <= 6; ++d) {
        float base = stm * ctd;
        mrow[row++] = cm * base;
        if (m > 0) mrow[row++] = sm * base;
        ctd *= ct;
      }
    }
    mrow[49] = 0.0f; mrow[50] = 0.0f; mrow[51] = 0.0f; // K padding
  }
  __syncthreads();

  // ---- cbf GEMM: [32 x 52] x [52 x 64] with V_WMMA_F32_16X16X4_F32 ----
  v8f acc[2][4] = {}; // [rowTile][nTile], C/D 16x16 f32 = 8 VGPRs each
  #pragma unroll
  for (int k0 = 0; k0 < KPAD; k0 += 4) {
    const int kk = k0 + 2 * hi; // A: lanes 0-15 K={k0,k0+1}; lanes 16-31 K={k0+2,k0+3}
    v2f a0 = *(const v2f*)(buf + (0 * 16 + l16) * KPAD + kk);
    v2f a1 = *(const v2f*)(buf + (1 * 16 + l16) * KPAD + kk);
    #pragma unroll
    for (int nt = 0; nt < 4; ++nt) {
      // B: lane-halves split K range; Wt is K-contiguous so one 8B load per frag
      v2f b = *(const v2f*)(ws + (size_t)(nt * 16 + l16) * KPAD + kk);
      acc[0][nt] = __builtin_amdgcn_wmma_f32_16x16x4_f32(
          false, a0, false, b, (short)0, acc[0][nt], false, false);
      acc[1][nt] = __builtin_amdgcn_wmma_f32_16x16x4_f32(
          false, a1, false, b, (short)0, acc[1][nt], false, false);
    }
  }
  __syncthreads(); // all A-fragment reads done before cbf overwrites the buffer

  // ---- scatter D tiles back to LDS: cbf[triplet][col] ----
  #pragma unroll
  for (int r = 0; r < 2; ++r) {
    const int mbase = r * 16 + 8 * hi; // C/D layout: lane half hi holds M = e + 8*hi
    #pragma unroll
    for (int nt = 0; nt < 4; ++nt) {
      const int col = nt * 16 + l16;
      if (col < NCBF) {
        #pragma unroll
        for (int e = 0; e < 8; ++e)
          buf[(mbase + e) * KPAD + col] = acc[r][nt][e];
      }
    }
  }
  __syncthreads();

  // ---- rbf in registers (recompute beats gathering a 42 MB table) ----
  const int   eidx = idx_kj[tc];
  const float x    = dist[eidx] * 0.2f; // 1/CUTOFF
  const float* norm = ws + NPAD * KPAD;
  float rbf[7][6];
  #pragma unroll
  for (int l = 0; l < 7; ++l)
    #pragma unroll
    for (int i = 0; i < 6; ++i)
      rbf[l][i] = norm[l * 6 + i] * sph_jn(l, x * Zc[l][i]);

  // ---- expansion + non-temporal streaming stores (588 MB, don't pollute L2) ----
  if (t < NUM_TRIPLETS) {
    float* op = out + (size_t)t * OUT_COLS;
    const float* crow = buf + lane * KPAD;
    #pragma unroll
    for (int a = 0; a < 7; ++a) {
      #pragma unroll
      for (int b = 0; b < 7; ++b) {
        const float c = crow[a * 7 + b];
        #pragma unroll
        for (int i = 0; i < 6; i += 2) {
          v2f v;
          v.x = c * rbf[b][i];
          v.y = c * rbf[b][i + 1];
          __builtin_nontemporal_store(v, (v2f*)(op + a * 42 + b * 6 + i));
        }
      }
    }
  }
}

extern "C" void kernel_launch(void* const* d_in, const int* in_sizes, int n_in,
                              void* d_out, int out_size, void* d_ws, size_t ws_size,
                              hipStream_t stream) {
  (void)in_sizes; (void)n_in; (void)out_size; (void)ws_size;
  const float* dist  = (const float*)d_in[0];
  const float* angle = (const float*)d_in[1];
  const float* phi   = (const float*)d_in[2];
  const int*   idx   = (const int*)d_in[3];
  float* ws  = (float*)d_ws;
  float* out = (float*)d_out;

  hipLaunchKernelGGL(torsion_init, dim3(1), dim3(256), 0, stream, ws);
  const int blocks = (NUM_TRIPLETS + 255) / 256; // 32 triplets/wave, 8 waves/block
  hipLaunchKernelGGL(torsion_main, dim3(blocks), dim3(256), 0, stream,
                     dist, angle, phi, idx, ws, out);
}